// GL_GINConv_3l_128h_44753559224362
// MI455X (gfx1250) — compile-verified
//
#include <hip/hip_runtime.h>
#include <math.h>

#define N_NODES 100000
#define N_EDGES 1600000
#define H       128
#define NC      10
#define NL      3
#define TILE_M  32    // rows per block (two 16-row WMMA sub-tiles per wave)
#define LDSS    132   // padded LDS row stride: 132 % 64 = 4 -> 16 rows hit 16 distinct banks

typedef float v2f __attribute__((ext_vector_type(2)));
typedef float v8f __attribute__((ext_vector_type(8)));

__device__ __forceinline__ float elu1(float x) {
    return x > 0.0f ? x : expm1f(x);
}

// ---------------------------------------------------------------------------
// Edge scatter: agg[dst] += h[src].  One wave32 per edge; each lane owns 4
// contiguous floats (global_load_b128 + 4x global_atomic_add_f32).
// ---------------------------------------------------------------------------
__global__ __launch_bounds__(256) void gin_scatter_kernel(
    const float* __restrict__ h, const int* __restrict__ src,
    const int* __restrict__ dst, float* __restrict__ agg) {
    const int gid  = blockIdx.x * blockDim.x + threadIdx.x;
    const int edge = gid >> 5;
    const int lane = gid & 31;
    if (edge >= N_EDGES) return;                // whole-wave uniform
    const int s = src[edge];
    const int d = dst[edge];
    const float4 v = *reinterpret_cast<const float4*>(h + (size_t)s * H + lane * 4);
    float* out = agg + (size_t)d * H + lane * 4;
    atomicAdd(out + 0, v.x);
    atomicAdd(out + 1, v.y);
    atomicAdd(out + 2, v.z);
    atomicAdd(out + 3, v.w);
}

// ---------------------------------------------------------------------------
// Fused GIN MLP for one layer: hout = elu(elu((hin+agg) @ Wa + ba) @ Wb + bb)
// One block = 32 nodes. 8 waves; wave w owns output columns [16w, 16w+16) and
// computes TWO 16x16 output tiles (rows 0-15 and 16-31) per V_WMMA_F32_16X16X4
// K-step, so each B fragment (weights, L2-hot) is reused twice.
// A fragment (16x4 f32): lane L -> M = L%16, K = 2*(L/16) + v  (ds_load_b64)
// B fragment (4x16 f32): lane L -> N = L%16, K = 2*(L/16) + v  (global)
// C/D tile:              lane L, vgpr v -> M = v + 8*(L/16), N = L%16
// ---------------------------------------------------------------------------
__global__ __launch_bounds__(256) void gin_mlp_kernel(
    const float* __restrict__ hin, const float* __restrict__ agg,
    const float* __restrict__ Wa, const float* __restrict__ ba,
    const float* __restrict__ Wb, const float* __restrict__ bb,
    float* __restrict__ hout) {
    __shared__ float xt[TILE_M * LDSS];   // 32*132*4 = 16.9 KB
    __shared__ float zt[TILE_M * LDSS];   // total 33.8 KB

    const int    tid     = threadIdx.x;
    const size_t rowBase = (size_t)blockIdx.x * TILE_M;

    // Warm L2/L1 for the weights while we stage the tile.
    __builtin_prefetch(Wa + (tid << 4), 0, 3);
    __builtin_prefetch(Wb + (tid << 4), 0, 3);

    // Cooperative vectorized load of the 32x128 activation tile, fusing h+agg.
    // 32*128/4 = 1024 float4s, 4 per thread; stride-132 keeps 16B alignment.
    for (int idx = tid; idx < TILE_M * (H / 4); idx += 256) {
        const int r  = idx >> 5;           // / (H/4)
        const int c4 = (idx & 31) << 2;    // column (float index)
        const size_t g = (rowBase + r) * H + c4;
        const float4 a = *reinterpret_cast<const float4*>(hin + g);
        const float4 b = *reinterpret_cast<const float4*>(agg + g);
        float4 s;
        s.x = a.x + b.x; s.y = a.y + b.y; s.z = a.z + b.z; s.w = a.w + b.w;
        *reinterpret_cast<float4*>(&xt[r * LDSS + c4]) = s;
    }
    __syncthreads();

    const int wave = tid >> 5;
    const int lane = tid & 31;
    const int mrow = lane & 15;          // A-row / output-col index within tile
    const int kgrp = (lane >> 4) << 1;   // 0 or 2
    const int hiM  = (lane >> 4) << 3;   // 0 or 8
    const int ncol = wave * 16 + mrow;   // global output column for B/C/D

    // ---- GEMM 1: z = elu(x @ Wa + ba) ----
    v8f acc0 = {0.f, 0.f, 0.f, 0.f, 0.f, 0.f, 0.f, 0.f};
    v8f acc1 = {0.f, 0.f, 0.f, 0.f, 0.f, 0.f, 0.f, 0.f};
#pragma unroll 4
    for (int k0 = 0; k0 < H; k0 += 4) {
        const v2f a0 = *reinterpret_cast<const v2f*>(&xt[mrow * LDSS + k0 + kgrp]);
        const v2f a1 = *reinterpret_cast<const v2f*>(&xt[(mrow + 16) * LDSS + k0 + kgrp]);
        v2f b;
        b.x = Wa[(k0 + kgrp + 0) * H + ncol];
        b.y = Wa[(k0 + kgrp + 1) * H + ncol];
        acc0 = __builtin_amdgcn_wmma_f32_16x16x4_f32(false, a0, false, b,
                                                     (short)0, acc0, false, false);
        acc1 = __builtin_amdgcn_wmma_f32_16x16x4_f32(false, a1, false, b,
                                                     (short)0, acc1, false, false);
    }
    const float biasA = ba[ncol];
#pragma unroll
    for (int v = 0; v < 8; ++v) {
        const int m = v + hiM;
        zt[m * LDSS + ncol]        = elu1(acc0[v] + biasA);
        zt[(m + 16) * LDSS + ncol] = elu1(acc1[v] + biasA);
    }
    __syncthreads();

    // ---- GEMM 2: h' = elu(z @ Wb + bb) ----
    v8f acc2 = {0.f, 0.f, 0.f, 0.f, 0.f, 0.f, 0.f, 0.f};
    v8f acc3 = {0.f, 0.f, 0.f, 0.f, 0.f, 0.f, 0.f, 0.f};
#pragma unroll 4
    for (int k0 = 0; k0 < H; k0 += 4) {
        const v2f a0 = *reinterpret_cast<const v2f*>(&zt[mrow * LDSS + k0 + kgrp]);
        const v2f a1 = *reinterpret_cast<const v2f*>(&zt[(mrow + 16) * LDSS + k0 + kgrp]);
        v2f b;
        b.x = Wb[(k0 + kgrp + 0) * H + ncol];
        b.y = Wb[(k0 + kgrp + 1) * H + ncol];
        acc2 = __builtin_amdgcn_wmma_f32_16x16x4_f32(false, a0, false, b,
                                                     (short)0, acc2, false, false);
        acc3 = __builtin_amdgcn_wmma_f32_16x16x4_f32(false, a1, false, b,
                                                     (short)0, acc3, false, false);
    }
    const float biasB = bb[ncol];
#pragma unroll
    for (int v = 0; v < 8; ++v) {
        const int m = v + hiM;
        hout[(rowBase + m) * H + ncol]      = elu1(acc2[v] + biasB);
        hout[(rowBase + m + 16) * H + ncol] = elu1(acc3[v] + biasB);
    }
}

// ---------------------------------------------------------------------------
// Classifier: out = h @ fcW + fcb.  N=10 is too narrow for a WMMA tile, so
// one wave per node: lane holds 4 of the 128 features, shuffle-reduce 10 dots.
// ---------------------------------------------------------------------------
__global__ __launch_bounds__(256) void gin_fc_kernel(
    const float* __restrict__ h, const float* __restrict__ fcW,
    const float* __restrict__ fcb, float* __restrict__ out) {
    const int gid  = blockIdx.x * blockDim.x + threadIdx.x;
    const int node = gid >> 5;
    const int lane = gid & 31;
    if (node >= N_NODES) return;
    float hv[4];
#pragma unroll
    for (int j = 0; j < 4; ++j) hv[j] = h[(size_t)node * H + lane + 32 * j];
#pragma unroll
    for (int c = 0; c < NC; ++c) {
        float p = 0.f;
#pragma unroll
        for (int j = 0; j < 4; ++j) p += hv[j] * fcW[(lane + 32 * j) * NC + c];
#pragma unroll
        for (int off = 16; off > 0; off >>= 1) p += __shfl_xor(p, off, 32);
        if (lane == 0) out[(size_t)node * NC + c] = p + fcb[c];
    }
}

extern "C" void kernel_launch(void* const* d_in, const int* in_sizes, int n_in,
                              void* d_out, int out_size, void* d_ws, size_t ws_size,
                              hipStream_t stream) {
    (void)in_sizes; (void)n_in; (void)out_size; (void)ws_size;
    const float* x   = (const float*)d_in[0];
    const int*   ei  = (const int*)  d_in[1];  // [2, E]; edge_weight (d_in[2]) unused
    const float* Wa  = (const float*)d_in[3];  // [3,128,128]
    const float* ba  = (const float*)d_in[4];  // [3,128]
    const float* Wb  = (const float*)d_in[5];
    const float* bb  = (const float*)d_in[6];
    const float* fcW = (const float*)d_in[7];  // [128,10]
    const float* fcb = (const float*)d_in[8];  // [10]
    float* out = (float*)d_out;

    const int* src = ei;
    const int* dst = ei + N_EDGES;

    const size_t nh = (size_t)N_NODES * H;
    float* agg = (float*)d_ws;
    float* hA  = agg + nh;
    float* hB  = hA + nh;
    float* bufs[2] = {hA, hB};

    const float* hin = x;
    for (int l = 0; l < NL; ++l) {
        hipMemsetAsync(agg, 0, nh * sizeof(float), stream);

        const int scatterBlocks = (N_EDGES * 32) / 256;       // exact: 200000
        gin_scatter_kernel<<<scatterBlocks, 256, 0, stream>>>(hin, src, dst, agg);

        float* hout = bufs[l & 1];
        gin_mlp_kernel<<<N_NODES / TILE_M, 256, 0, stream>>>( // exact: 3125
            hin, agg,
            Wa + (size_t)l * H * H, ba + (size_t)l * H,
            Wb + (size_t)l * H * H, bb + (size_t)l * H,
            hout);
        hin = hout;
    }

    const int fcBlocks = (N_NODES * 32 + 255) / 256;          // 12500
    gin_fc_kernel<<<fcBlocks, 256, 0, stream>>>(hin, fcW, fcb, out);
}